// GCN4_9294309228814
// MI455X (gfx1250) — compile-verified
//
#include <hip/hip_runtime.h>
#include <hip/hip_bf16.h>

typedef float v2f __attribute__((ext_vector_type(2)));
typedef float v8f __attribute__((ext_vector_type(8)));

#define NN    50000   // nodes
#define NE    800000  // edges
#define NF    7       // input features
#define NC    5       // classes
#define NG    64      // graphs
#define HID   128     // hidden width

// ---------------------------------------------------------------------------
// Utility: zero a float buffer (needed each call; harness does not re-poison).
// ---------------------------------------------------------------------------
__global__ void zero_f32(float* __restrict__ p, int n) {
    int i = blockIdx.x * blockDim.x + threadIdx.x;
    if (i < n) p[i] = 0.0f;
}

// ---------------------------------------------------------------------------
// deg[dst] += ew  (edge-weight degree, atomics into zeroed buffer)
// ---------------------------------------------------------------------------
__global__ void deg_kernel(const int* __restrict__ dst,
                           const float* __restrict__ ew,
                           float* __restrict__ deg) {
    int e = blockIdx.x * blockDim.x + threadIdx.x;
    if (e < NE) atomicAdd(&deg[dst[e]], ew[e]);
}

// dinv[i] = rsqrt(deg[i] + 1)   (in place; deg >= 1 always since ew >= 0)
__global__ void dinv_kernel(float* __restrict__ deg) {
    int i = blockIdx.x * blockDim.x + threadIdx.x;
    if (i < NN) deg[i] = rsqrtf(deg[i] + 1.0f);
}

// ---------------------------------------------------------------------------
// C[M x HID] = A[M x K] * B[K x HID] using V_WMMA_F32_16X16X4_F32.
// Block = 256 threads = 8 wave32s; block handles 16 rows, each wave one
// 16-column tile. K stepped by 4 (zero-padded when K % 4 != 0).
//
// VGPR layouts per CDNA5 ISA 7.12.2:
//   A 16x4 : lane<16 -> {K=k0,k0+1}, lane>=16 -> {K=k0+2,k0+3}, M = lane&15
//   B 4x16 : VGPR v, lane half h -> K = k0 + 2h + v, N = lane&15
//   C 16x16: VGPR j -> M = j + 8*(lane>>4), N = lane&15
// ---------------------------------------------------------------------------
template <int K>
__global__ void __launch_bounds__(256)
gemm_wmma(const float* __restrict__ A, const float* __restrict__ B,
          float* __restrict__ C) {
    const int lane = threadIdx.x & 31;
    const int wave = threadIdx.x >> 5;
    const int half = lane >> 4;
    const int lm   = lane & 15;
    const int row0 = blockIdx.x * 16;
    const int col  = wave * 16 + lm;      // output column (0..127)
    const int arow = row0 + lm;           // A row this lane supplies

    v8f acc = {};
    constexpr int KP = (K + 3) & ~3;
    for (int k0 = 0; k0 < KP; k0 += 4) {
        const int ka = k0 + 2 * half;
        v2f a, b;
        a.x = (ka     < K) ? A[(size_t)arow * K + ka]     : 0.0f;
        a.y = (ka + 1 < K) ? A[(size_t)arow * K + ka + 1] : 0.0f;
        b.x = (ka     < K) ? B[(size_t)(ka    ) * HID + col] : 0.0f;
        b.y = (ka + 1 < K) ? B[(size_t)(ka + 1) * HID + col] : 0.0f;
        acc = __builtin_amdgcn_wmma_f32_16x16x4_f32(
            /*neg_a=*/false, a, /*neg_b=*/false, b,
            /*c_mod=*/(short)0, acc, /*reuse_a=*/false, /*reuse_b=*/false);
    }
#pragma unroll
    for (int j = 0; j < 8; ++j)
        C[(size_t)(row0 + j + 8 * half) * HID + col] = acc[j];
}

// ---------------------------------------------------------------------------
// Edge scatter: agg[dst] += h[src] * (dinv[src]*ew*dinv[dst]).
// One wave per edge; each lane owns 4 consecutive features (128-bit gather,
// 4 scalar f32 atomics). Activations fit in L2 (192MB) so the random
// gather/RMW traffic should mostly hit L2.
// ---------------------------------------------------------------------------
__global__ void __launch_bounds__(256)
scatter_kernel(const float* __restrict__ h, const int* __restrict__ src,
               const int* __restrict__ dst, const float* __restrict__ ew,
               const float* __restrict__ dinv, float* __restrict__ agg) {
    const int lane = threadIdx.x & 31;
    const int e = blockIdx.x * 8 + (threadIdx.x >> 5);
    if (e >= NE) return;
    const int s = src[e];
    const int d = dst[e];
    const float coeff = dinv[s] * ew[e] * dinv[d];
    const float4 hv =
        *reinterpret_cast<const float4*>(&h[(size_t)s * HID + lane * 4]);
    float* out = &agg[(size_t)d * HID + lane * 4];
    atomicAdd(out + 0, hv.x * coeff);
    atomicAdd(out + 1, hv.y * coeff);
    atomicAdd(out + 2, hv.z * coeff);
    atomicAdd(out + 3, hv.w * coeff);
}

// ---------------------------------------------------------------------------
// out = agg + h * dinv^2 + bias, optional ReLU (in place on agg).
// ---------------------------------------------------------------------------
__global__ void combine_kernel(float* __restrict__ agg,
                               const float* __restrict__ h,
                               const float* __restrict__ dinv,
                               const float* __restrict__ bias, int relu) {
    int i = blockIdx.x * blockDim.x + threadIdx.x;
    if (i >= NN * HID) return;
    const int node = i >> 7;
    const int f    = i & (HID - 1);
    const float di = dinv[node];
    float v = agg[i] + h[i] * di * di + bias[f];
    if (relu) v = fmaxf(v, 0.0f);
    agg[i] = v;
}

// ---------------------------------------------------------------------------
// Mean pool: sums[batch[n]] += act[n]; counts[batch[n]] += 1.
// ---------------------------------------------------------------------------
__global__ void __launch_bounds__(256)
pool_sum_kernel(const float* __restrict__ act, const int* __restrict__ batch,
                float* __restrict__ sums, float* __restrict__ counts) {
    const int lane = threadIdx.x & 31;
    const int node = blockIdx.x * 8 + (threadIdx.x >> 5);
    if (node >= NN) return;
    const int g = batch[node];
    const float4 v =
        *reinterpret_cast<const float4*>(&act[(size_t)node * HID + lane * 4]);
    float* o = &sums[g * HID + lane * 4];
    atomicAdd(o + 0, v.x);
    atomicAdd(o + 1, v.y);
    atomicAdd(o + 2, v.z);
    atomicAdd(o + 3, v.w);
    if (lane == 0) atomicAdd(&counts[g], 1.0f);
}

// out[g,c] = (sums[g]/max(counts[g],1)) . Wl[:,c] + bl[c]   (64x5 outputs)
__global__ void head_kernel(const float* __restrict__ sums,
                            const float* __restrict__ counts,
                            const float* __restrict__ Wl,
                            const float* __restrict__ bl,
                            float* __restrict__ out) {
    int i = blockIdx.x * blockDim.x + threadIdx.x;
    if (i >= NG * NC) return;
    const int g = i / NC;
    const int c = i % NC;
    const float inv = 1.0f / fmaxf(counts[g], 1.0f);
    float acc = 0.0f;
    for (int f = 0; f < HID; ++f)
        acc += sums[g * HID + f] * Wl[f * NC + c];
    out[i] = acc * inv + bl[c];
}

// ---------------------------------------------------------------------------
extern "C" void kernel_launch(void* const* d_in, const int* in_sizes, int n_in,
                              void* d_out, int out_size, void* d_ws,
                              size_t ws_size, hipStream_t stream) {
    const float* x     = (const float*)d_in[0];
    const int*   ei    = (const int*)d_in[1];
    const int*   src   = ei;            // edge_index[0]
    const int*   dst   = ei + NE;       // edge_index[1]
    const float* ew    = (const float*)d_in[2];
    const int*   batch = (const int*)d_in[3];
    const float* W1 = (const float*)d_in[4];
    const float* b1 = (const float*)d_in[5];
    const float* W2 = (const float*)d_in[6];
    const float* b2 = (const float*)d_in[7];
    const float* W3 = (const float*)d_in[8];
    const float* b3 = (const float*)d_in[9];
    const float* Wl = (const float*)d_in[10];
    const float* bl = (const float*)d_in[11];
    float* out = (float*)d_out;

    // Workspace layout (floats): 3 activation buffers + dinv + pool scratch.
    float* ws     = (float*)d_ws;
    const size_t NH = (size_t)NN * HID;      // 6.4M floats
    float* bufH   = ws;                      // GEMM output h = act @ W
    float* bufA   = ws + NH;                 // agg / activation ping
    float* bufB   = ws + 2 * NH;             // agg / activation pong
    float* dinv   = ws + 3 * NH;             // NN floats (deg -> dinv in place)
    float* sums   = dinv + NN;               // NG*HID
    float* counts = sums + NG * HID;         // NG

    const int T = 256;
    const int NHb = (int)((NH + T - 1) / T);

    // --- degree + dinv (shared across all three layers) ---
    zero_f32<<<(NN + T - 1) / T, T, 0, stream>>>(dinv, NN);
    deg_kernel<<<(NE + T - 1) / T, T, 0, stream>>>(dst, ew, dinv);
    dinv_kernel<<<(NN + T - 1) / T, T, 0, stream>>>(dinv);

    // --- layer 1: x[NN x 7] @ W1 -> relu(gcn) into bufA ---
    gemm_wmma<NF><<<NN / 16, 256, 0, stream>>>(x, W1, bufH);
    zero_f32<<<NHb, T, 0, stream>>>(bufA, (int)NH);
    scatter_kernel<<<(NE + 7) / 8, 256, 0, stream>>>(bufH, src, dst, ew, dinv, bufA);
    combine_kernel<<<NHb, T, 0, stream>>>(bufA, bufH, dinv, b1, 1);

    // --- layer 2: bufA @ W2 -> relu(gcn) into bufB ---
    gemm_wmma<HID><<<NN / 16, 256, 0, stream>>>(bufA, W2, bufH);
    zero_f32<<<NHb, T, 0, stream>>>(bufB, (int)NH);
    scatter_kernel<<<(NE + 7) / 8, 256, 0, stream>>>(bufH, src, dst, ew, dinv, bufB);
    combine_kernel<<<NHb, T, 0, stream>>>(bufB, bufH, dinv, b2, 1);

    // --- layer 3: bufB @ W3 -> gcn (no relu) into bufA ---
    gemm_wmma<HID><<<NN / 16, 256, 0, stream>>>(bufB, W3, bufH);
    zero_f32<<<NHb, T, 0, stream>>>(bufA, (int)NH);
    scatter_kernel<<<(NE + 7) / 8, 256, 0, stream>>>(bufH, src, dst, ew, dinv, bufA);
    combine_kernel<<<NHb, T, 0, stream>>>(bufA, bufH, dinv, b3, 0);

    // --- global mean pool + linear head ---
    zero_f32<<<(NG * HID + T - 1) / T, T, 0, stream>>>(sums, NG * HID);
    zero_f32<<<1, T, 0, stream>>>(counts, NG);
    pool_sum_kernel<<<(NN + 7) / 8, 256, 0, stream>>>(bufA, batch, sums, counts);
    head_kernel<<<(NG * NC + T - 1) / T, T, 0, stream>>>(sums, counts, Wl, bl, out);
}